// CapsuleLayer_40664750358956
// MI455X (gfx1250) — compile-verified
//
#include <hip/hip_runtime.h>
#include <hip/hip_bf16.h>
#include <math.h>

// Capsule routing, fully fused-recompute strategy for MI455X (gfx1250).
// fp32 WMMA 16x16x4: A = X tile [16b x K], B = W^T tile [K x 16i], K=DI=8 -> 2 WMMAs.

typedef float v2f __attribute__((ext_vector_type(2)));
typedef float v8f __attribute__((ext_vector_type(8)));

#define BB   64
#define CCAP 32
#define NN   1152
#define DI   8
#define DC   16
#define ROUT 3

#define NCHUNK 64            // n-chunks per batch-tile -> 256 blocks of work
#define NPER   (NN / NCHUNK) // 18 capsules-in per block
#define BTILES (BB / 16)     // 4 batch tiles of 16 rows

// ---------------------------------------------------------------------------
__global__ void init_logits_kernel(float* __restrict__ logits, int n) {
    int i = blockIdx.x * blockDim.x + threadIdx.x;
    if (i < n) logits[i] = 0.0f;
}

// softmax over C for each (b,n); logits/coef layout [B][N][C]
__global__ void coef_kernel(const float* __restrict__ logits,
                            float* __restrict__ coef) {
    int t = blockIdx.x * blockDim.x + threadIdx.x; // t = b*NN + n
    if (t >= BB * NN) return;
    const float* lp = logits + (size_t)t * CCAP;
    float* cp = coef + (size_t)t * CCAP;
    float m = lp[0];
#pragma unroll
    for (int c = 1; c < CCAP; ++c) m = fmaxf(m, lp[c]);
    float e[CCAP];
    float s = 0.0f;
#pragma unroll
    for (int c = 0; c < CCAP; ++c) { e[c] = __expf(lp[c] - m); s += e[c]; }
    float inv = 1.0f / s;
#pragma unroll
    for (int c = 0; c < CCAP; ++c) cp[c] = e[c] * inv;
}

// ---------------------------------------------------------------------------
// s[b,c,i] = sum_n coef[b,n,c] * hat[b,c,n,i]
//          = sum_n (coef * X_row) x W^T  -> coef folded into WMMA A rows,
// WMMA C-operand accumulates s across the n loop directly.
// grid(NCHUNK, BTILES), 256 threads (8 waves); wave owns 4 capsules c.
__global__ void s_pass_kernel(const float* __restrict__ X,
                              const float* __restrict__ W,
                              const float* __restrict__ coef,
                              float* __restrict__ partials) {
    const int lane = threadIdx.x & 31;
    const int wave = threadIdx.x >> 5;
    const int row  = lane & 15;   // A: M-row / B: i-col / D: N-col
    const int hi   = lane >> 4;   // K-pair select (ISA 16x4 fp32 layout)
    const int bbase = blockIdx.y * 16;
    const int nbase = blockIdx.x * NPER;
    const int c0 = wave * 4;

    v8f acc[4] = {};

    for (int nn = 0; nn < NPER; ++nn) {
        const int n = nbase + nn;
        const float* xp = X + ((size_t)(bbase + row) * NN + n) * DI + hi * 2;
        v2f a0 = { xp[0], xp[1] };   // K = hi*2 + {0,1}
        v2f a1 = { xp[4], xp[5] };   // K = 4 + hi*2 + {0,1}
        const float* cfp = coef + ((size_t)(bbase + row) * NN + n) * CCAP + c0;
#pragma unroll
        for (int q = 0; q < 4; ++q) {
            const int c = c0 + q;
            const float cf = cfp[q];
            v2f as0 = a0 * cf;
            v2f as1 = a1 * cf;
            const float* wp = W + (((size_t)c * NN + n) * DC + row) * DI + hi * 2;
            v2f b0 = { wp[0], wp[1] };
            v2f b1 = { wp[4], wp[5] };
            acc[q] = __builtin_amdgcn_wmma_f32_16x16x4_f32(
                false, as0, false, b0, (short)0, acc[q], false, false);
            acc[q] = __builtin_amdgcn_wmma_f32_16x16x4_f32(
                false, as1, false, b1, (short)0, acc[q], false, false);
        }
    }
    // D layout: lane l, vgpr r -> (m = r + hi*8, i = l&15)
#pragma unroll
    for (int q = 0; q < 4; ++q) {
        const int c = c0 + q;
#pragma unroll
        for (int r = 0; r < 8; ++r) {
            const int b = bbase + r + hi * 8;
            partials[(((size_t)blockIdx.x * BB + b) * CCAP + c) * DC + row] = acc[q][r];
        }
    }
}

// reduce n-chunk partials and apply squash; one thread per (b,c)
__global__ void squash_kernel(const float* __restrict__ partials,
                              float* __restrict__ out) {
    int t = blockIdx.x * blockDim.x + threadIdx.x; // t = b*CCAP + c
    if (t >= BB * CCAP) return;
    float v[DC];
#pragma unroll
    for (int i = 0; i < DC; ++i) v[i] = 0.0f;
    for (int p = 0; p < NCHUNK; ++p) {
        const float* pp = partials + ((size_t)p * BB * CCAP + t) * DC;
#pragma unroll
        for (int i = 0; i < DC; ++i) v[i] += pp[i];
    }
    float sq = 0.0f;
#pragma unroll
    for (int i = 0; i < DC; ++i) sq += v[i] * v[i];
    float scale = sq / ((1.0f + sq) * sqrtf(sq + 1e-7f));
    float* op = out + (size_t)t * DC;
#pragma unroll
    for (int i = 0; i < DC; ++i) op[i] = scale * v[i];
}

// ---------------------------------------------------------------------------
// logits[b,n,c] += sum_i out[b,c,i] * hat[b,c,n,i]
// hat tile recomputed with WMMA (C=0), then row-wise dot over the i-lane dim
// via shfl_xor butterflies (masks < 16 stay within each wave32 half).
__global__ void agree_kernel(const float* __restrict__ X,
                             const float* __restrict__ W,
                             const float* __restrict__ out,
                             float* __restrict__ logits) {
    const int lane = threadIdx.x & 31;
    const int wave = threadIdx.x >> 5;
    const int row  = lane & 15;
    const int hi   = lane >> 4;
    const int bbase = blockIdx.y * 16;
    const int nbase = blockIdx.x * NPER;
    const int c0 = wave * 4;

    // outputs are n-invariant: preload ov[q][r] = out[b(m), c, i(lane)]
    float ov[4][8];
#pragma unroll
    for (int q = 0; q < 4; ++q)
#pragma unroll
        for (int r = 0; r < 8; ++r)
            ov[q][r] = out[(((size_t)(bbase + r + hi * 8)) * CCAP + (c0 + q)) * DC + row];

    for (int nn = 0; nn < NPER; ++nn) {
        const int n = nbase + nn;
        const float* xp = X + ((size_t)(bbase + row) * NN + n) * DI + hi * 2;
        v2f a0 = { xp[0], xp[1] };
        v2f a1 = { xp[4], xp[5] };
#pragma unroll
        for (int q = 0; q < 4; ++q) {
            const int c = c0 + q;
            const float* wp = W + (((size_t)c * NN + n) * DC + row) * DI + hi * 2;
            v2f b0 = { wp[0], wp[1] };
            v2f b1 = { wp[4], wp[5] };
            v8f d = {};
            d = __builtin_amdgcn_wmma_f32_16x16x4_f32(
                false, a0, false, b0, (short)0, d, false, false);
            d = __builtin_amdgcn_wmma_f32_16x16x4_f32(
                false, a1, false, b1, (short)0, d, false, false);
#pragma unroll
            for (int r = 0; r < 8; ++r) {
                float p = d[r] * ov[q][r];
                p += __shfl_xor(p, 1, 32);
                p += __shfl_xor(p, 2, 32);
                p += __shfl_xor(p, 4, 32);
                p += __shfl_xor(p, 8, 32);
                if (row == 0) {
                    const int b = bbase + r + hi * 8;
                    logits[((size_t)b * NN + n) * CCAP + c] += p; // unique writer
                }
            }
        }
    }
}

// ---------------------------------------------------------------------------
extern "C" void kernel_launch(void* const* d_in, const int* in_sizes, int n_in,
                              void* d_out, int out_size, void* d_ws, size_t ws_size,
                              hipStream_t stream) {
    (void)in_sizes; (void)n_in; (void)out_size; (void)ws_size;
    const float* X = (const float*)d_in[0];   // [B,N,DI]
    const float* W = (const float*)d_in[1];   // [C,N,DC,DI]
    float* out = (float*)d_out;               // [B,C,DC]

    float* logits   = (float*)d_ws;                              // B*N*C
    float* coef     = logits   + (size_t)BB * NN * CCAP;         // B*N*C
    float* partials = coef     + (size_t)BB * NN * CCAP;         // NCHUNK*B*C*DC
    float* outbuf   = partials + (size_t)NCHUNK * BB * CCAP * DC;// B*C*DC

    const int nlog = BB * NN * CCAP;
    init_logits_kernel<<<(nlog + 255) / 256, 256, 0, stream>>>(logits, nlog);

    for (int t = 0; t < ROUT; ++t) {
        coef_kernel<<<(BB * NN + 255) / 256, 256, 0, stream>>>(logits, coef);
        s_pass_kernel<<<dim3(NCHUNK, BTILES), 256, 0, stream>>>(X, W, coef, partials);
        float* o = (t == ROUT - 1) ? out : outbuf;
        squash_kernel<<<(BB * CCAP + 255) / 256, 256, 0, stream>>>(partials, o);
        if (t < ROUT - 1)
            agree_kernel<<<dim3(NCHUNK, BTILES), 256, 0, stream>>>(X, W, o, logits);
    }
}